// ConditionalFeedForward_2018634629557
// MI455X (gfx1250) — compile-verified
//
#include <hip/hip_runtime.h>
#include <cstdint>

// Problem dims (fixed by the reference)
#define T_TOK 1024
#define DDIM  1024
#define HDIM  2048
#define NEXP  32
#define TOPK  2
#define NPAIR (T_TOK * TOPK)

typedef __attribute__((ext_vector_type(2))) float v2f;
typedef __attribute__((ext_vector_type(8))) float v8f;

// 16-byte payload type for the async LDS copy builtin
// (param0 is int4 in AS(1), param1 is int4 in AS(3)).
typedef int v4i __attribute__((__vector_size__(16)));
typedef __attribute__((address_space(1))) v4i glb_v4i;
typedef __attribute__((address_space(3))) v4i lds_v4i;

#if defined(__has_builtin)
#if __has_builtin(__builtin_amdgcn_global_load_async_to_lds_b128)
#define HAVE_ASYNC_LDS 1
#endif
#endif
#ifndef HAVE_ASYNC_LDS
#define HAVE_ASYNC_LDS 0
#endif

// One 16-byte HBM -> LDS copy per active lane (GLOBAL_LOAD_ASYNC_TO_LDS_B128).
__device__ __forceinline__ void async_ld16(void* lds, const void* g) {
#if HAVE_ASYNC_LDS
  __builtin_amdgcn_global_load_async_to_lds_b128(
      (glb_v4i*)(uintptr_t)g, (lds_v4i*)(uint32_t)(uintptr_t)lds,
      /*offset=*/0, /*cpol=*/0);
#else
  *(float4*)lds = *(const float4*)g;
#endif
}

__device__ __forceinline__ void async_wait0() {
#if HAVE_ASYNC_LDS
#if __has_builtin(__builtin_amdgcn_s_wait_asynccnt)
  __builtin_amdgcn_s_wait_asynccnt(0);
#else
  asm volatile("s_wait_asynccnt 0x0" ::: "memory");
#endif
#endif
}

// D(16x16,f32) = A(16x4,f32) x B(4x16,f32) + C
__device__ __forceinline__ v8f wmma4(v2f a, v2f b, v8f c) {
  return __builtin_amdgcn_wmma_f32_16x16x4_f32(
      /*neg_a=*/false, a, /*neg_b=*/false, b,
      /*c_mod=*/(short)0, c, /*reuse_a=*/false, /*reuse_b=*/false);
}

// fast SiLU: x * sigmoid(x), sigmoid via v_rcp_f32
__device__ __forceinline__ float fast_silu(float g) {
  return g * __builtin_amdgcn_rcpf(1.0f + __expf(-g));
}

// ---------------- routing ----------------
__global__ void k_zero_counts(int* counts) {
  if (threadIdx.x < NEXP) counts[threadIdx.x] = 0;
}

__global__ void k_route(const int* __restrict__ eidx,
                        int* __restrict__ counts,
                        int* __restrict__ lists) {
  int i = blockIdx.x * blockDim.x + threadIdx.x;
  if (i < NPAIR) {
    int e = eidx[i];
    int p = atomicAdd(&counts[e], 1);
    lists[e * NPAIR + p] = i;
  }
}

// ---------------- gate/up GEMM + SiLU ----------------
// grid: (HDIM/128, NEXP), block: 256 (8 waves, each owns a 16x16 WMMA tile)
// K-chunk = 16, double-buffered async HBM->LDS staging.
#define GU_DK 16
#define GU_LDW 20  // padded row stride (floats), 16B-aligned rows, bank-spread
__global__ void __launch_bounds__(256) k_gateup(
    const float* __restrict__ x, const float* __restrict__ Wgu,
    const int* __restrict__ counts, const int* __restrict__ lists,
    float* __restrict__ inter) {
  const int e   = blockIdx.y;
  const int h0  = blockIdx.x * 128;
  const int cnt = counts[e];
  if (cnt == 0) return;
  const int* lst = lists + e * NPAIR;

  __shared__ __align__(16) float sX[2][16][GU_LDW];
  __shared__ __align__(16) float sWg[2][128][GU_LDW];
  __shared__ __align__(16) float sWu[2][128][GU_LDW];
  __shared__ int rowI[16];
  __shared__ int rowT[16];

  const int tid  = threadIdx.x;
  const int lane = tid & 31;
  const int wv   = tid >> 5;
  const int l15  = lane & 15;
  const int kh   = (lane >> 4) << 1;   // K offset 0 (lanes 0-15) / 2 (lanes 16-31)
  const int nrow = (wv << 4) + l15;    // local output column 0..127

  // Per-thread loader geometry (invariant across the whole kernel).
  const int wr0 = tid >> 2;            // W row for p=0 (0..63)
  const int wc  = (tid & 3) << 2;      // W col offset (0,4,8,12)
  // W row source pointers: fixed for this block.
  const float* wg0 = Wgu + (((size_t)e * 2 + 0) * HDIM + h0 + wr0) * DDIM + wc;
  const float* wg1 = wg0 + (size_t)64 * DDIM;
  const float* wu0 = Wgu + (((size_t)e * 2 + 1) * HDIM + h0 + wr0) * DDIM + wc;
  const float* wu1 = wu0 + (size_t)64 * DDIM;

  for (int rb = 0; rb < cnt; rb += 16) {
    if (tid < 16) {
      int r = rb + tid;
      int i = lst[r < cnt ? r : cnt - 1];
      rowI[tid] = i;
      rowT[tid] = i >> 1;  // TOPK == 2 -> token id
    }
    __syncthreads();

    // Hoist the gathered X row pointer into a register for this row-tile.
    const float* xrow = (tid < 64)
        ? x + (size_t)rowT[tid >> 2] * DDIM + ((tid & 3) << 2)
        : nullptr;

    auto load_tiles = [&](int dc, int buf) {
      if (tid < 64) {
        async_ld16(&sX[buf][tid >> 2][(tid & 3) << 2], xrow + dc);
      }
      async_ld16(&sWg[buf][wr0][wc],      wg0 + dc);
      async_ld16(&sWg[buf][wr0 + 64][wc], wg1 + dc);
      async_ld16(&sWu[buf][wr0][wc],      wu0 + dc);
      async_ld16(&sWu[buf][wr0 + 64][wc], wu1 + dc);
    };

    v8f ag = {0.f, 0.f, 0.f, 0.f, 0.f, 0.f, 0.f, 0.f};
    v8f au = {0.f, 0.f, 0.f, 0.f, 0.f, 0.f, 0.f, 0.f};

    load_tiles(0, 0);           // prologue
    async_wait0();
    __syncthreads();

    int cur = 0;
    for (int dc = 0; dc < DDIM; dc += GU_DK) {
      const int nxt = cur ^ 1;
      if (dc + GU_DK < DDIM) load_tiles(dc + GU_DK, nxt);  // overlap with WMMA

#pragma unroll
      for (int kk = 0; kk < GU_DK / 4; ++kk) {
        const int kb = (kk << 2) + kh;
        v2f a  = *(const v2f*)&sX[cur][l15][kb];
        v2f bg = *(const v2f*)&sWg[cur][nrow][kb];
        v2f bu = *(const v2f*)&sWu[cur][nrow][kb];
        ag = wmma4(a, bg, ag);
        au = wmma4(a, bu, au);
      }

      async_wait0();
      __syncthreads();
      cur = nxt;
    }

    // fused SiLU(gate) * up -> inter[pair][h]
    const int n  = h0 + nrow;
    const int mb = (lane >> 4) << 3;
#pragma unroll
    for (int v = 0; v < 8; ++v) {
      const int m = mb + v;
      if (rb + m < cnt) {
        inter[(size_t)rowI[m] * HDIM + n] = fast_silu(ag[v]) * au[v];
      }
    }
    __syncthreads();
  }
}

// ---------------- down GEMM + scatter ----------------
// grid: (DDIM/128, NEXP), block: 256. K-chunk = 32, double-buffered.
#define DN_DK 32
#define DN_LDW 36
__global__ void __launch_bounds__(256) k_down(
    const float* __restrict__ inter, const float* __restrict__ Wd,
    const int* __restrict__ counts, const int* __restrict__ lists,
    float* __restrict__ out) {
  const int e   = blockIdx.y;
  const int d0  = blockIdx.x * 128;
  const int cnt = counts[e];
  if (cnt == 0) return;
  const int* lst = lists + e * NPAIR;

  __shared__ __align__(16) float sA[2][16][DN_LDW];
  __shared__ __align__(16) float sW[2][128][DN_LDW];
  __shared__ int rowI[16];

  const int tid  = threadIdx.x;
  const int lane = tid & 31;
  const int wv   = tid >> 5;
  const int l15  = lane & 15;
  const int kh   = (lane >> 4) << 1;
  const int nrow = (wv << 4) + l15;

  // Per-thread loader geometry.
  const int wr0 = tid >> 3;            // 0..31
  const int wc  = (tid & 7) << 2;      // 0..28
  const float* wp = Wd + ((size_t)e * DDIM + d0 + wr0) * HDIM + wc;

  for (int rb = 0; rb < cnt; rb += 16) {
    if (tid < 16) {
      int r = rb + tid;
      rowI[tid] = lst[r < cnt ? r : cnt - 1];
    }
    __syncthreads();

    const float* arow = (tid < 128)
        ? inter + (size_t)rowI[tid >> 3] * HDIM + ((tid & 7) << 2)
        : nullptr;

    auto load_tiles = [&](int hc, int buf) {
      if (tid < 128) {
        async_ld16(&sA[buf][tid >> 3][(tid & 7) << 2], arow + hc);
      }
#pragma unroll
      for (int p = 0; p < 4; ++p) {
        async_ld16(&sW[buf][wr0 + p * 32][wc], wp + (size_t)(p * 32) * HDIM + hc);
      }
    };

    v8f acc = {0.f, 0.f, 0.f, 0.f, 0.f, 0.f, 0.f, 0.f};

    load_tiles(0, 0);           // prologue
    async_wait0();
    __syncthreads();

    int cur = 0;
    for (int hc = 0; hc < HDIM; hc += DN_DK) {
      const int nxt = cur ^ 1;
      if (hc + DN_DK < HDIM) load_tiles(hc + DN_DK, nxt);

#pragma unroll
      for (int kk = 0; kk < DN_DK / 4; ++kk) {
        const int kb = (kk << 2) + kh;
        v2f a = *(const v2f*)&sA[cur][l15][kb];
        v2f b = *(const v2f*)&sW[cur][nrow][kb];
        acc = wmma4(a, b, acc);
      }

      async_wait0();
      __syncthreads();
      cur = nxt;
    }

    // scatter to out[(t,k), d] == out[pair * DDIM + d]
    const int n  = d0 + nrow;
    const int mb = (lane >> 4) << 3;
#pragma unroll
    for (int v = 0; v < 8; ++v) {
      const int m = mb + v;
      if (rb + m < cnt) {
        out[(size_t)rowI[m] * DDIM + n] = acc[v];
      }
    }
    __syncthreads();
  }
}

// ---------------- launch ----------------
extern "C" void kernel_launch(void* const* d_in, const int* in_sizes, int n_in,
                              void* d_out, int out_size, void* d_ws, size_t ws_size,
                              hipStream_t stream) {
  (void)in_sizes; (void)n_in; (void)out_size; (void)ws_size;

  const float* x    = (const float*)d_in[0];
  const int*   eidx = (const int*)d_in[1];
  const float* Wgu  = (const float*)d_in[2];
  const float* Wd   = (const float*)d_in[3];
  float*       out  = (float*)d_out;

  // workspace layout: counts[64] | lists[E*NPAIR] | inter[NPAIR*HDIM]
  int*   counts = (int*)d_ws;
  int*   lists  = counts + 64;
  float* inter  = (float*)(void*)(lists + NEXP * NPAIR);

  hipLaunchKernelGGL(k_zero_counts, dim3(1), dim3(64), 0, stream, counts);
  hipLaunchKernelGGL(k_route, dim3(NPAIR / 256), dim3(256), 0, stream,
                     eidx, counts, lists);
  hipLaunchKernelGGL(k_gateup, dim3(HDIM / 128, NEXP), dim3(256), 0, stream,
                     x, Wgu, counts, lists, inter);
  hipLaunchKernelGGL(k_down, dim3(DDIM / 128, NEXP), dim3(256), 0, stream,
                     inter, Wd, counts, lists, out);
}